// GNN_19464791785636
// MI455X (gfx1250) — compile-verified
//
#include <hip/hip_runtime.h>
#include <math.h>

typedef __attribute__((ext_vector_type(2))) float v2f;
typedef __attribute__((ext_vector_type(8))) float v8f;

#define D_FEAT   128
#define HIDDEN   16
#define NCLASS   40

// ---------------------------------------------------------------------------
// Kernel 1: zero the aggregation buffers (agg_h[N*16] ++ deg[N], contiguous)
// ---------------------------------------------------------------------------
__global__ void gnn_zero_kernel(float* __restrict__ p, long long n) {
    long long i = (long long)blockIdx.x * blockDim.x + threadIdx.x;
    if (i < n) p[i] = 0.0f;
}

// ---------------------------------------------------------------------------
// Kernel 2: projection  y = x @ W_l, xr = x @ W_r  using fp32 WMMA.
// One wave per 16-node tile. K=128 in steps of 4 (V_WMMA_F32_16X16X4_F32).
// A fragment (16x4 f32): lanes 0-15 hold row M=lane, K={k0,k0+1};
//                        lanes 16-31 hold row M=lane-16, K={k0+2,k0+3}.
// B fragment (4x16 f32): lanes 0-15 hold col N=lane, K rows {k0,k0+1};
//                        lanes 16-31 hold col N=lane-16, K rows {k0+2,k0+3}.
// C/D (16x16 f32): VGPR i -> row (i + 8*(lane>=16)), col lane%16.
// ---------------------------------------------------------------------------
__global__ void gnn_proj_kernel(const float* __restrict__ x,
                                const float* __restrict__ Wl,
                                const float* __restrict__ Wr,
                                float* __restrict__ y,
                                float* __restrict__ xr,
                                int nTiles) {
    __shared__ float sWl[D_FEAT * HIDDEN];   // 8 KB
    __shared__ float sWr[D_FEAT * HIDDEN];   // 8 KB

    const int tid = threadIdx.x;
    for (int i = tid; i < D_FEAT * HIDDEN; i += 256) {
        sWl[i] = Wl[i];
        sWr[i] = Wr[i];
    }
    __syncthreads();

    const int wave = tid >> 5;            // 8 waves / block
    const int lane = tid & 31;
    const int tile = blockIdx.x * 8 + wave;
    if (tile >= nTiles) return;           // wave-uniform branch: EXEC all-1 for WMMA

    const int m16   = lane & 15;          // row within tile (A) / col (B,C,D)
    const int khalf = (lane >> 4) << 1;   // 0 or 2: K sub-pair selector

    const float* __restrict__ xrow = x + (size_t)(tile * 16 + m16) * D_FEAT;

    v8f cl = {};
    v8f cr = {};

    #pragma unroll 4
    for (int k0 = 0; k0 < D_FEAT; k0 += 4) {
        const int ka = k0 + khalf;
        v2f a;                                   // A fragment (global, b64 load)
        a.x = xrow[ka];
        a.y = xrow[ka + 1];
        v2f bl, br;                              // B fragments from LDS
        bl.x = sWl[ka * HIDDEN + m16];
        bl.y = sWl[(ka + 1) * HIDDEN + m16];
        br.x = sWr[ka * HIDDEN + m16];
        br.y = sWr[(ka + 1) * HIDDEN + m16];
        cl = __builtin_amdgcn_wmma_f32_16x16x4_f32(false, a, false, bl,
                                                   (short)0, cl, false, false);
        cr = __builtin_amdgcn_wmma_f32_16x16x4_f32(false, a, false, br,
                                                   (short)0, cr, false, false);
    }

    // Store D (C-layout): VGPR i -> row tile*16 + i + 8*(lane>=16), col m16.
    const int rhalf = (lane >> 4) << 3;   // 0 or 8
    #pragma unroll
    for (int i = 0; i < 8; ++i) {
        const size_t m = (size_t)tile * 16 + rhalf + i;
        y [m * HIDDEN + m16] = cl[i];
        xr[m * HIDDEN + m16] = cr[i];
    }
}

// ---------------------------------------------------------------------------
// Kernel 3: edge scatter in projected (16-dim) space.
// 16 threads per edge: lane f adds y[src][f] into agg[dst][f]; f==0 bumps deg.
// ---------------------------------------------------------------------------
__global__ void gnn_edge_kernel(const int* __restrict__ src,
                                const int* __restrict__ dst,
                                const float* __restrict__ y,
                                float* __restrict__ agg,
                                float* __restrict__ deg,
                                int E) {
    const long long idx = (long long)blockIdx.x * blockDim.x + threadIdx.x;
    const long long e = idx >> 4;
    const int f = (int)(idx & 15);
    if (e >= E) return;
    const int s = src[e];
    const int d = dst[e];
    atomicAdd(&agg[(size_t)d * HIDDEN + f], y[(size_t)s * HIDDEN + f]);
    if (f == 0) atomicAdd(&deg[d], 1.0f);
}

// ---------------------------------------------------------------------------
// Kernel 4: finalize. One thread per node:
//   h = relu(agg/max(deg,1) + b_l + xr);  logits = h@W3 + b3;  log_softmax.
// W3/b3/b_l staged in LDS.
// ---------------------------------------------------------------------------
__global__ void gnn_finalize_kernel(const float* __restrict__ agg,
                                    const float* __restrict__ deg,
                                    const float* __restrict__ xr,
                                    const float* __restrict__ b_l,
                                    const float* __restrict__ W3,
                                    const float* __restrict__ b3,
                                    float* __restrict__ out,
                                    int N) {
    __shared__ float sW3[HIDDEN * NCLASS];   // 2.5 KB
    __shared__ float sb3[NCLASS];
    __shared__ float sbl[HIDDEN];

    const int tid = threadIdx.x;
    for (int i = tid; i < HIDDEN * NCLASS; i += 256) sW3[i] = W3[i];
    if (tid < NCLASS) sb3[tid] = b3[tid];
    if (tid < HIDDEN) sbl[tid] = b_l[tid];
    __syncthreads();

    const int n = blockIdx.x * blockDim.x + tid;
    if (n >= N) return;

    const float inv = 1.0f / fmaxf(deg[n], 1.0f);
    const float* ap = agg + (size_t)n * HIDDEN;
    const float* xp = xr  + (size_t)n * HIDDEN;

    float h[HIDDEN];
    #pragma unroll
    for (int j = 0; j < HIDDEN; ++j)
        h[j] = fmaxf(fmaf(ap[j], inv, sbl[j]) + xp[j], 0.0f);

    float lg[NCLASS];
    float mx = -INFINITY;
    #pragma unroll
    for (int c = 0; c < NCLASS; ++c) {
        float s = sb3[c];
        #pragma unroll
        for (int j = 0; j < HIDDEN; ++j)
            s = fmaf(h[j], sW3[j * NCLASS + c], s);
        lg[c] = s;
        mx = fmaxf(mx, s);
    }

    float sum = 0.0f;
    #pragma unroll
    for (int c = 0; c < NCLASS; ++c) sum += __expf(lg[c] - mx);
    const float lse = __logf(sum);

    float* op = out + (size_t)n * NCLASS;
    #pragma unroll
    for (int c = 0; c < NCLASS; ++c) op[c] = lg[c] - mx - lse;
}

// ---------------------------------------------------------------------------
// Host-side orchestration (graph-capture safe: kernels only, all on stream).
// Workspace layout: agg[N*16] | deg[N] | y[N*16] | xr[N*16]   (~19.6 MB)
// ---------------------------------------------------------------------------
extern "C" void kernel_launch(void* const* d_in, const int* in_sizes, int n_in,
                              void* d_out, int out_size, void* d_ws, size_t ws_size,
                              hipStream_t stream) {
    const float* x   = (const float*)d_in[0];
    const int*   ei  = (const int*)  d_in[1];
    const float* Wl  = (const float*)d_in[2];
    const float* b_l = (const float*)d_in[3];
    const float* Wr  = (const float*)d_in[4];
    const float* W3  = (const float*)d_in[5];
    const float* b3  = (const float*)d_in[6];
    float* out = (float*)d_out;

    const int N = in_sizes[0] / D_FEAT;   // 100000
    const int E = in_sizes[1] / 2;        // 3200000

    float* agg = (float*)d_ws;
    float* deg = agg + (size_t)N * HIDDEN;
    float* y   = deg + N;
    float* xr  = y + (size_t)N * HIDDEN;

    // 1) zero agg + deg (contiguous: N*17 floats)
    {
        const long long nz = (long long)N * (HIDDEN + 1);
        const int grid = (int)((nz + 255) / 256);
        gnn_zero_kernel<<<grid, 256, 0, stream>>>(agg, nz);
    }
    // 2) WMMA projection: y = x@Wl, xr = x@Wr
    {
        const int nTiles = (N + 15) / 16;           // 6250
        const int grid = (nTiles + 7) / 8;
        gnn_proj_kernel<<<grid, 256, 0, stream>>>(x, Wl, Wr, y, xr, nTiles);
    }
    // 3) edge scatter (16 threads / edge)
    {
        const long long t = (long long)E * HIDDEN;
        const int grid = (int)((t + 255) / 256);
        gnn_edge_kernel<<<grid, 256, 0, stream>>>(ei, ei + E, y, agg, deg, E);
    }
    // 4) finalize + log_softmax
    {
        const int grid = (N + 255) / 256;
        gnn_finalize_kernel<<<grid, 256, 0, stream>>>(agg, deg, xr, b_l, W3, b3,
                                                      out, N);
    }
}